// LaplaceLoss_2611340116548
// MI455X (gfx1250) — compile-verified
//
#include <hip/hip_runtime.h>
#include <hip/hip_bf16.h>
#include <math.h>

typedef __attribute__((ext_vector_type(2))) float v2f;
typedef __attribute__((ext_vector_type(8))) float v8f;

#define MC 2048
#define NC 2048
#define DF 256
#define NFULL 50000
#define KCORR 4096
#define NBACK 30000
#define MAXP 256

// ---- workspace layout (bytes) ----
#define OFF_FD     0                          // 2048*2048*4 = 16777216
#define OFF_DEDUP  16777216                   // 524288 (2048*2048 bits)
#define OFF_FGT    (OFF_DEDUP + 524288)       // 524288 final corr_gt bitmask
#define OFF_BMR    (OFF_FGT   + 524288)       // 6400  (50000-bit bitmap, ref)
#define OFF_BMS    (OFF_BMR   + 6400)         // 6400  (src)
#define OFF_CTR    (OFF_BMS   + 6400)         // 256   (count + float accums)
#define OFF_IDXR   (OFF_CTR   + 256)          // 8192
#define OFF_IDXS   (OFF_IDXR  + 8192)
#define OFF_MSKR   (OFF_IDXS  + 8192)
#define OFF_MSKS   (OFF_MSKR  + 8192)
#define OFF_CVAL   (OFF_MSKS  + 8192)         // 16384
#define OFF_CIDX   (OFF_CVAL  + 16384)        // 16384
#define ZERO_WORDS ((524288*2 + 6400*2 + 256)/4)

// ---------------- zero init ----------------
__global__ void k_zero(unsigned* p, int n) {
    int i = blockIdx.x * 256 + threadIdx.x;
    if (i < n) p[i] = 0u;
}

// ------------- back-index bitmaps -------------
__global__ void k_bitmap(const int* ref_back, const int* src_back,
                         unsigned* bmr, unsigned* bms) {
    int t = blockIdx.x * 256 + threadIdx.x;
    if (t < NBACK) {
        int v = ref_back[t];
        atomicOr(&bmr[v >> 5], 1u << (v & 31));
    } else if (t < 2 * NBACK) {
        int v = src_back[t - NBACK];
        atomicOr(&bms[v >> 5], 1u << (v & 31));
    }
}

// ------------- NN argmin (one block per corr point) -------------
__global__ void k_nn(const float* ref_full, const float* src_full,
                     const float* ref_c, const float* src_c,
                     const unsigned* bmr, const unsigned* bms,
                     int* idxr, int* idxs, int* mskr, int* msks) {
    int b = blockIdx.x, t = threadIdx.x;
    const float* full;
    const float* cp;
    if (b < MC) { full = ref_full; cp = ref_c + 3 * b; }
    else        { full = src_full; cp = src_c + 3 * (b - MC); }
    float cx = cp[0], cy = cp[1], cz = cp[2];
    float cn = cx * cx + cy * cy + cz * cz;
    float best = 3.0e38f; int bi = 0;
    for (int j = t; j < NFULL; j += 256) {
        float px = full[3 * j], py = full[3 * j + 1], pz = full[3 * j + 2];
        float d = cn + (px * px + py * py + pz * pz)
                     - 2.0f * (cx * px + cy * py + cz * pz);
        if (d < best) { best = d; bi = j; }
    }
    __shared__ float sd[256];
    __shared__ int   si[256];
    sd[t] = best; si[t] = bi;
    __syncthreads();
    for (int s = 128; s > 0; s >>= 1) {
        if (t < s) {
            float o = sd[t + s]; int oi = si[t + s];
            if (o < sd[t] || (o == sd[t] && oi < si[t])) { sd[t] = o; si[t] = oi; }
        }
        __syncthreads();
    }
    if (t == 0) {
        int w = si[0];
        if (b < MC) { idxr[b] = w;      mskr[b]      = (bmr[w >> 5] >> (w & 31)) & 1; }
        else        { idxs[b - MC] = w; msks[b - MC] = (bms[w >> 5] >> (w & 31)) & 1; }
    }
}

// ------------- gt scatter + dedup + candidate list -------------
__global__ void k_scatter(const int* gt_idx, const float* overlaps,
                          const int* mskr, const int* msks,
                          unsigned* dedup, float* cval, int* cidx, int* cnt) {
    int k = blockIdx.x * 256 + threadIdx.x;
    if (k >= KCORR) return;
    int xi = gt_idx[2 * k], yi = gt_idx[2 * k + 1];
    if (mskr[xi] && msks[yi]) {
        int flat = xi * NC + yi;
        unsigned bit = 1u << (flat & 31);
        unsigned old = atomicOr(&dedup[flat >> 5], bit);
        if (!(old & bit)) {
            int pos = atomicAdd(cnt, 1);
            cval[pos] = overlaps[k];
            cidx[pos] = flat;
        }
    }
}

// ------------- top-256 selection (single block, bitonic) -------------
__global__ void k_select(const float* cval, const int* cidx, const int* cnt,
                         unsigned* fgt) {
    __shared__ float sv[4096];
    __shared__ int   sx[4096];
    int t = threadIdx.x;
    int n = *cnt;
    if (n <= MAXP) {
        for (int i = t; i < n; i += 256) {
            int f = cidx[i];
            atomicOr(&fgt[f >> 5], 1u << (f & 31));
        }
        return;
    }
    for (int i = t; i < 4096; i += 256) {
        if (i < n) { sv[i] = cval[i]; sx[i] = cidx[i]; }
        else       { sv[i] = -3.0e38f; sx[i] = 0x7FFFFFFF; }
    }
    __syncthreads();
    for (int k = 2; k <= 4096; k <<= 1) {
        for (int j = k >> 1; j > 0; j >>= 1) {
            for (int i = t; i < 4096; i += 256) {
                int ixj = i ^ j;
                if (ixj > i) {
                    bool desc = ((i & k) == 0);
                    float va = sv[i], vb = sv[ixj];
                    int   ia = sx[i], ib = sx[ixj];
                    // "b should come before a" under (value desc, index asc)
                    bool b_first = (vb > va) || (vb == va && ib < ia);
                    if (b_first == desc) {
                        sv[i] = vb; sv[ixj] = va;
                        sx[i] = ib; sx[ixj] = ia;
                    }
                }
            }
            __syncthreads();
        }
    }
    if (t < MAXP && sv[t] > -0.5f) {
        int f = sx[t];
        atomicOr(&fgt[f >> 5], 1u << (f & 31));
    }
}

// ------------- WMMA GEMM + feat_dists (one wave = 16x64 strip) -------------
__global__ void k_gemm(const float* __restrict__ ref_feats,
                       const float* __restrict__ src_feats,
                       const float* __restrict__ sp_mask,
                       float* __restrict__ fd) {
    int w       = blockIdx.x * 8 + (threadIdx.x >> 5); // 4096 waves
    int lane    = threadIdx.x & 31;
    int half    = lane >> 4;
    int r       = lane & 15;
    int rowTile = w >> 5;   // 0..127
    int colStrip = w & 31;  // 0..31
    int aRow  = rowTile * 16 + r;
    int cBase = colStrip * 64;
    const float* aPtr = ref_feats + aRow * DF;

    v8f acc[4] = {};
    for (int k0 = 0; k0 < DF; k0 += 4) {
        int kk = k0 + 2 * half;
        v2f a = *(const v2f*)(aPtr + kk);
#pragma unroll
        for (int cc = 0; cc < 4; ++cc) {
            const float* bPtr = src_feats + (cBase + cc * 16 + r) * DF + kk;
            v2f b = *(const v2f*)bPtr;
            acc[cc] = __builtin_amdgcn_wmma_f32_16x16x4_f32(
                false, a, false, b, (short)0, acc[cc], false, false);
        }
    }
#pragma unroll
    for (int cc = 0; cc < 4; ++cc) {
        int C = cBase + cc * 16 + r;
        float vms = 1.0f - sp_mask[MC + C];
#pragma unroll
        for (int rr = 0; rr < 8; ++rr) {
            int R = rowTile * 16 + rr + 8 * half;
            float vmr = 1.0f - sp_mask[R];
            float s   = acc[cc][rr];
            float fsq = fmaxf(2.0f - 2.0f * s, 0.0f);
            fd[R * NC + C] = expf(0.5f * vmr * vms) * sqrtf(fsq);
        }
    }
}

__device__ __forceinline__ float softplus_stable(float z) {
    return (z > 0.0f) ? z + log1pf(expf(-z)) : log1pf(expf(z));
}

// ------------- per-row logsumexp loss (block per row) -------------
__global__ void k_row(const float* __restrict__ fd, const unsigned* __restrict__ fgt,
                      float* acc_row) {
    __shared__ float slp[NC];
    __shared__ float sln[NC];
    __shared__ float red[256];
    int i = blockIdx.x, t = threadIdx.x;
    for (int j = t; j < NC; j += 256) {
        int flat = i * NC + j;
        float v = fd[flat];
        unsigned g = (fgt[flat >> 5] >> (flat & 31)) & 1u;
        float tp = fmaxf(v - 0.1f, 0.0f);
        float tn = fmaxf(1.4f - v, 0.0f);
        slp[j] = g ? 24.0f * tp * tp : 0.0f;
        sln[j] = g ? 0.0f : 24.0f * tn * tn;
    }
    __syncthreads();
    float mlp = -3.0e38f, mln = -3.0e38f;
    for (int j = t; j < NC; j += 256) { mlp = fmaxf(mlp, slp[j]); mln = fmaxf(mln, sln[j]); }
    red[t] = mlp; __syncthreads();
    for (int s = 128; s > 0; s >>= 1) { if (t < s) red[t] = fmaxf(red[t], red[t + s]); __syncthreads(); }
    float Mlp = red[0]; __syncthreads();
    red[t] = mln; __syncthreads();
    for (int s = 128; s > 0; s >>= 1) { if (t < s) red[t] = fmaxf(red[t], red[t + s]); __syncthreads(); }
    float Mln = red[0]; __syncthreads();
    float sl = 0.0f, sn = 0.0f;
    for (int j = t; j < NC; j += 256) { sl += expf(slp[j] - Mlp); sn += expf(sln[j] - Mln); }
    red[t] = sl; __syncthreads();
    for (int s = 128; s > 0; s >>= 1) { if (t < s) red[t] += red[t + s]; __syncthreads(); }
    float Slp = red[0]; __syncthreads();
    red[t] = sn; __syncthreads();
    for (int s = 128; s > 0; s >>= 1) { if (t < s) red[t] += red[t + s]; __syncthreads(); }
    float Sln = red[0];
    if (t == 0) {
        float z = (Mlp + logf(Slp)) + (Mln + logf(Sln));
        atomicAdd(acc_row, softplus_stable(z) / 24.0f);
    }
}

// ------------- per-col online logsumexp loss (thread per col) -------------
__global__ void k_col(const float* __restrict__ fd, const unsigned* __restrict__ fgt,
                      float* acc_col) {
    int t = threadIdx.x;
    int col = blockIdx.x * 256 + t;
    float mlp = -3.0e38f, slp = 0.0f, mln = -3.0e38f, sln = 0.0f;
    for (int i = 0; i < MC; ++i) {
        int flat = i * NC + col;
        float v = fd[flat];
        unsigned g = (fgt[flat >> 5] >> (flat & 31)) & 1u;
        float tp = fmaxf(v - 0.1f, 0.0f);
        float tn = fmaxf(1.4f - v, 0.0f);
        float lp = g ? 24.0f * tp * tp : 0.0f;
        float ln = g ? 0.0f : 24.0f * tn * tn;
        if (lp > mlp) { slp = slp * expf(mlp - lp) + 1.0f; mlp = lp; } else slp += expf(lp - mlp);
        if (ln > mln) { sln = sln * expf(mln - ln) + 1.0f; mln = ln; } else sln += expf(ln - mln);
    }
    float z = (mlp + logf(slp)) + (mln + logf(sln));
    float val = softplus_stable(z) / 24.0f;
    __shared__ float red[256];
    red[t] = val; __syncthreads();
    for (int s = 128; s > 0; s >>= 1) { if (t < s) red[t] += red[t + s]; __syncthreads(); }
    if (t == 0) atomicAdd(acc_col, red[0]);
}

// ------------- loss2 = mean |var_mask - gt_mask| -------------
__global__ void k_loss2(const float* sp_mask, const int* mskr, const int* msks,
                        float* acc_l2) {
    int t = threadIdx.x;
    int i = blockIdx.x * 256 + t;
    float v = 0.0f;
    if (i < MC + NC) {
        float gtm = (i < MC) ? (float)mskr[i] : (float)msks[i - MC];
        v = fabsf((1.0f - sp_mask[i]) - gtm);
    }
    __shared__ float red[256];
    red[t] = v; __syncthreads();
    for (int s = 128; s > 0; s >>= 1) { if (t < s) red[t] += red[t + s]; __syncthreads(); }
    if (t == 0) atomicAdd(acc_l2, red[0]);
}

// ------------- combine -------------
__global__ void k_final(const float* acc_row, const float* acc_col,
                        const float* acc_l2, float* out) {
    if (threadIdx.x == 0 && blockIdx.x == 0) {
        float l1 = ((*acc_row) / (float)MC + (*acc_col) / (float)NC) * 0.5f;
        float l2 = (*acc_l2) / (float)(MC + NC);
        out[0] = l1 + l2;
        out[1] = l1;
        out[2] = l2;
    }
}

extern "C" void kernel_launch(void* const* d_in, const int* in_sizes, int n_in,
                              void* d_out, int out_size, void* d_ws, size_t ws_size,
                              hipStream_t stream) {
    const float* ref_points   = (const float*)d_in[0];
    const float* src_points   = (const float*)d_in[1];
    const float* ref_points_c = (const float*)d_in[2];
    const float* src_points_c = (const float*)d_in[3];
    const float* ref_feats_c  = (const float*)d_in[4];
    const float* src_feats_c  = (const float*)d_in[5];
    const int*   gt_idx       = (const int*)d_in[6];
    const float* gt_overlaps  = (const float*)d_in[7];
    const float* sp_mask      = (const float*)d_in[8];
    const int*   ref_back     = (const int*)d_in[9];
    const int*   src_back     = (const int*)d_in[10];

    unsigned char* ws = (unsigned char*)d_ws;
    float*    fd    = (float*)(ws + OFF_FD);
    unsigned* dedup = (unsigned*)(ws + OFF_DEDUP);
    unsigned* fgt   = (unsigned*)(ws + OFF_FGT);
    unsigned* bmr   = (unsigned*)(ws + OFF_BMR);
    unsigned* bms   = (unsigned*)(ws + OFF_BMS);
    int*      cnt   = (int*)(ws + OFF_CTR);
    float*    accs  = (float*)(ws + OFF_CTR + 16); // [0]=row [1]=col [2]=l2
    int*      idxr  = (int*)(ws + OFF_IDXR);
    int*      idxs  = (int*)(ws + OFF_IDXS);
    int*      mskr  = (int*)(ws + OFF_MSKR);
    int*      msks  = (int*)(ws + OFF_MSKS);
    float*    cval  = (float*)(ws + OFF_CVAL);
    int*      cidx  = (int*)(ws + OFF_CIDX);
    float*    out   = (float*)d_out;

    // 1) zero bitmasks / counters
    k_zero<<<(ZERO_WORDS + 255) / 256, 256, 0, stream>>>(dedup, ZERO_WORDS);
    // 2) back-index bitmaps
    k_bitmap<<<(2 * NBACK + 255) / 256, 256, 0, stream>>>(ref_back, src_back, bmr, bms);
    // 3) NN argmin + masks (blocks 0..2047 ref, 2048..4095 src)
    k_nn<<<MC + NC, 256, 0, stream>>>(ref_points, src_points, ref_points_c, src_points_c,
                                      bmr, bms, idxr, idxs, mskr, msks);
    // 4) gt scatter with dedup
    k_scatter<<<(KCORR + 255) / 256, 256, 0, stream>>>(gt_idx, gt_overlaps, mskr, msks,
                                                       dedup, cval, cidx, cnt);
    // 5) top-256 cap -> final gt bitmask
    k_select<<<1, 256, 0, stream>>>(cval, cidx, cnt, fgt);
    // 6) WMMA GEMM + feat_dists (4096 waves = 512 blocks x 8 waves)
    k_gemm<<<512, 256, 0, stream>>>(ref_feats_c, src_feats_c, sp_mask, fd);
    // 7) row / col losses
    k_row<<<MC, 256, 0, stream>>>(fd, fgt, &accs[0]);
    k_col<<<NC / 256, 256, 0, stream>>>(fd, fgt, &accs[1]);
    // 8) loss2
    k_loss2<<<(MC + NC) / 256, 256, 0, stream>>>(sp_mask, mskr, msks, &accs[2]);
    // 9) combine
    k_final<<<1, 32, 0, stream>>>(&accs[0], &accs[1], &accs[2], out);
}